// Net_18803366822520
// MI455X (gfx1250) — compile-verified
//
#include <hip/hip_runtime.h>

#define CH   64
#define GS   100
#define K1   (GS * CH)   // 6400

typedef __attribute__((ext_vector_type(2))) float v2f;
typedef __attribute__((ext_vector_type(8))) float v8f;
typedef __attribute__((ext_vector_type(4))) int   v4i;

#if defined(__has_builtin)
#  if __has_builtin(__builtin_amdgcn_global_load_async_to_lds_b128)
#    define HAVE_ASYNC_LDS 1
#  else
#    define HAVE_ASYNC_LDS 0
#  endif
#else
#  define HAVE_ASYNC_LDS 0
#endif

#if HAVE_ASYNC_LDS
typedef __attribute__((address_space(1))) v4i gv4i;   // global (__device__) int4
typedef __attribute__((address_space(3))) v4i lv4i;   // LDS (__shared__) int4
#endif

__device__ __forceinline__ v8f wmma_f32(v2f a, v2f b, v8f c) {
    // D = A(16x4) * B(4x16) + C(16x16), exact f32 accumulate
    return __builtin_amdgcn_wmma_f32_16x16x4_f32(
        /*neg_a=*/false, a, /*neg_b=*/false, b,
        /*c_mod=*/(short)0, c, /*reuse_a=*/false, /*reuse_b=*/false);
}

__device__ __forceinline__ v2f ld2(const float* p) { return *(const v2f*)p; }

// ---------------------------------------------------------------------------
// Stage 0a/0b: repack W_d1 [6400,256] and W_d2 [256,128] into K-pair
// interleaved layout: Wp[p][c] = (W[2p][c], W[2p+1][c]) so a WMMA B operand
// is one aligned b64 load straight into an even VGPR pair.
// ---------------------------------------------------------------------------
__global__ void __launch_bounds__(256)
repack_pairs(const float* __restrict__ W, float* __restrict__ Wp, int ncols) {
    const int s = blockIdx.x * blockDim.x + threadIdx.x;   // one float2 slot
    const int p = s / ncols, c = s % ncols;
    v2f w;
    w.x = W[(size_t)(2 * p) * ncols + c];
    w.y = W[(size_t)(2 * p + 1) * ncols + c];
    *(v2f*)(Wp + (size_t)s * 2) = w;
}

// ---------------------------------------------------------------------------
// Stage 1: edge scatter-add.  One wave per edge (edge id is wave-uniform so
// index loads scalarize); each lane owns 2 channels -> 2x global_atomic_add_f32
// into the L2-resident (105 MB < 192 MB L2) agg buffer.
// ---------------------------------------------------------------------------
__global__ void __launch_bounds__(256)
edge_scatter(const float* __restrict__ x,
             const int* __restrict__ snd,
             const int* __restrict__ rcv,
             float* __restrict__ agg, int nE) {
    const int lane = threadIdx.x & 31;
    const int wave = threadIdx.x >> 5;
    const int wavesPerGrid = gridDim.x * (blockDim.x >> 5);
    for (int e = blockIdx.x * (blockDim.x >> 5) + wave; e < nE; e += wavesPerGrid) {
        const int s = snd[e];
        const int r = rcv[e];
        const float2 v = ((const float2*)(x + (size_t)s * CH))[lane];
        float* dst = agg + (size_t)r * CH + 2 * lane;
        unsafeAtomicAdd(dst,     v.x);
        unsafeAtomicAdd(dst + 1, v.y);
    }
}

// ---------------------------------------------------------------------------
// Stage 2: h = ELU(BN(agg @ W1 + b1)) in place.  W1 staged into LDS in the
// K-pair interleaved layout; per-wave 16x64 A tile in LDS; 16 K-chunks x 4
// N-tiles of V_WMMA_F32_16X16X4_F32 with pure b64 operand loads.
// ---------------------------------------------------------------------------
__global__ void __launch_bounds__(256)
gemm1_bn_elu(float* __restrict__ h,
             const float* __restrict__ W1, const float* __restrict__ b1,
             const float* __restrict__ gamma, const float* __restrict__ beta,
             const float* __restrict__ bn_mean, const float* __restrict__ bn_var) {
    __shared__ float ldsW2[CH * CH];         // [p:32][n:64][2] interleaved, 16 KB
    __shared__ float ldsA[8 * 16 * CH];      // per-wave 16x64 tiles, 32 KB
    __shared__ float sScale[CH], sShift[CH];

    const int tid = threadIdx.x;
    #pragma unroll
    for (int i = 0; i < 8; ++i) {            // interleave W1 into LDS
        const int slot = i * 256 + tid;      // 0..2047
        const int p = slot >> 6, c = slot & 63;
        v2f w;
        w.x = W1[(2 * p) * CH + c];
        w.y = W1[(2 * p + 1) * CH + c];
        *(v2f*)(ldsW2 + slot * 2) = w;
    }
    if (tid < CH) {
        const float s = gamma[tid] * rsqrtf(bn_var[tid] + 1e-3f);
        sScale[tid] = s;
        sShift[tid] = (b1[tid] - bn_mean[tid]) * s + beta[tid];
    }
    __syncthreads();

    const int wave = tid >> 5, lane = tid & 31;
    const int hl = lane >> 4;                // half-wave
    const int n  = lane & 15;
    const int koff = hl * 2;
    const size_t rowBase = (size_t)(blockIdx.x * 8 + wave) * 16;
    float* ldsa = ldsA + wave * (16 * CH);

    const float* A = h + rowBase * CH;       // stage 16x64 A tile
    #pragma unroll
    for (int i = 0; i < 8; ++i) {
        const int slot = i * 32 + lane;
        ((float4*)ldsa)[slot] = ((const float4*)A)[slot];
    }

    v8f acc0 = {}, acc1 = {}, acc2 = {}, acc3 = {};
    #pragma unroll
    for (int kc = 0; kc < 16; ++kc) {
        const v2f a = ld2(ldsa + n * CH + kc * 4 + koff);        // A: row n, K pair
        const float* wrow = ldsW2 + (size_t)(2 * kc + hl) * 128; // interleaved pair row
        acc0 = wmma_f32(a, ld2(wrow + (     n) * 2), acc0);
        acc1 = wmma_f32(a, ld2(wrow + (16 + n) * 2), acc1);
        acc2 = wmma_f32(a, ld2(wrow + (32 + n) * 2), acc2);
        acc3 = wmma_f32(a, ld2(wrow + (48 + n) * 2), acc3);
    }

    float* Hrow = h + rowBase * CH;
    #pragma unroll
    for (int j = 0; j < 4; ++j) {
        const v8f acc = (j == 0) ? acc0 : (j == 1) ? acc1 : (j == 2) ? acc2 : acc3;
        const int col = j * 16 + n;
        const float s = sScale[col], sh = sShift[col];
        #pragma unroll
        for (int v = 0; v < 8; ++v) {
            float val = acc[v] * s + sh;
            val = val > 0.f ? val : (__expf(val) - 1.f);   // ELU
            Hrow[(hl * 8 + v) * CH + col] = val;
        }
    }
}

// ---------------------------------------------------------------------------
// Stage 3: Y = ReLU(X[4096,6400] @ W_d1 + b) using the repacked Wp.
// Block = 64 rows x 64 cols (4 waves); 100 K-panels of 64; the interleaved
// 16 KB B panel is staged with GLOBAL_LOAD_ASYNC_TO_LDS_B128 when available.
// ---------------------------------------------------------------------------
__global__ void __launch_bounds__(128)
dense1_relu(const float* __restrict__ X, const float* __restrict__ Wp,
            const float* __restrict__ bias, float* __restrict__ Y) {
    __shared__ float ldsB2[32 * 128];        // 32 k-pairs x 64 cols x 2, 16 KB
    const int tid = threadIdx.x;
    const int wave = tid >> 5, lane = tid & 31;
    const int hl = lane >> 4, n = lane & 15, koff = hl * 2;
    const int c0 = blockIdx.y * 64;
    const int rowBase = blockIdx.x * 64 + wave * 16;

    v8f acc0 = {}, acc1 = {}, acc2 = {}, acc3 = {};
    for (int kp = 0; kp < K1; kp += 64) {
        const int kpHalf = kp >> 1;          // pair-row index into Wp
        __syncthreads();
#if HAVE_ASYNC_LDS
        #pragma unroll
        for (int i = 0; i < 8; ++i) {
            const int slot = i * 128 + tid;          // 1024 x 16B = 16 KB
            const int pl = slot >> 5, cg = slot & 31;
            const float* g = Wp + (size_t)(kpHalf + pl) * 512 + c0 * 2 + cg * 4;
            float* l = ldsB2 + slot * 4;
            __builtin_amdgcn_global_load_async_to_lds_b128((gv4i*)g, (lv4i*)l, 0, 0);
        }
#  if __has_builtin(__builtin_amdgcn_s_wait_asynccnt)
        __builtin_amdgcn_s_wait_asynccnt(0);
#  else
        asm volatile("s_wait_asynccnt 0x0" ::: "memory");
#  endif
#else
        #pragma unroll
        for (int i = 0; i < 8; ++i) {
            const int slot = i * 128 + tid;
            const int pl = slot >> 5, cg = slot & 31;
            *(float4*)(ldsB2 + slot * 4) =
                *(const float4*)(Wp + (size_t)(kpHalf + pl) * 512 + c0 * 2 + cg * 4);
        }
#endif
        if (kp + 64 < K1 && tid < 32)        // -> global_prefetch_b8 (next panel)
            __builtin_prefetch(Wp + (size_t)(kpHalf + 32 + tid) * 512 + c0 * 2, 0, 1);
        __syncthreads();

        const float* Xr = X + (size_t)(rowBase + n) * K1 + kp + koff;
        #pragma unroll
        for (int sc = 0; sc < 16; ++sc) {
            const v2f a = ld2(Xr + sc * 4);                      // global b64
            const float* brow = ldsB2 + (size_t)(2 * sc + hl) * 128;
            acc0 = wmma_f32(a, ld2(brow + (     n) * 2), acc0);  // ds b64 each
            acc1 = wmma_f32(a, ld2(brow + (16 + n) * 2), acc1);
            acc2 = wmma_f32(a, ld2(brow + (32 + n) * 2), acc2);
            acc3 = wmma_f32(a, ld2(brow + (48 + n) * 2), acc3);
        }
    }

    float* Yr = Y + (size_t)rowBase * 256 + c0;
    #pragma unroll
    for (int j = 0; j < 4; ++j) {
        const v8f acc = (j == 0) ? acc0 : (j == 1) ? acc1 : (j == 2) ? acc2 : acc3;
        const int col = j * 16 + n;
        const float bb = bias[c0 + col];
        #pragma unroll
        for (int v = 0; v < 8; ++v) {
            const float val = acc[v] + bb;
            Yr[(hl * 8 + v) * 256 + col] = val > 0.f ? val : 0.f;
        }
    }
}

// ---------------------------------------------------------------------------
// Stage 4: Y = ReLU(X[4096,256] @ W_d2 + b) from the repacked Wp2 (L2 resident,
// B operand is one global b64 per WMMA).  Per-wave 16x16 tile.
// ---------------------------------------------------------------------------
__global__ void __launch_bounds__(256)
dense2_relu(const float* __restrict__ X, const float* __restrict__ Wp2,
            const float* __restrict__ bias, float* __restrict__ Y) {
    const int tid = threadIdx.x;
    const int wave = tid >> 5, lane = tid & 31;
    const int hl = lane >> 4, n = lane & 15, koff = hl * 2;
    const int wid = blockIdx.x * 8 + wave;
    const int rowBase = (wid >> 3) * 16;
    const int colBase = (wid & 7) * 16;

    v8f acc = {};
    const float* Xr = X + (size_t)(rowBase + n) * 256 + koff;
    const float* Wr = Wp2 + (size_t)(colBase + n) * 2;
    #pragma unroll 8
    for (int sc = 0; sc < 64; ++sc) {
        const v2f a = ld2(Xr + sc * 4);
        const v2f b = ld2(Wr + (size_t)(2 * sc + hl) * 256);
        acc = wmma_f32(a, b, acc);
    }
    const float bb = bias[colBase + n];
    #pragma unroll
    for (int v = 0; v < 8; ++v) {
        const float val = acc[v] + bb;
        Y[(size_t)(rowBase + hl * 8 + v) * 128 + colBase + n] = val > 0.f ? val : 0.f;
    }
}

// ---------------------------------------------------------------------------
// Stage 5: out = sigmoid(X[4096,128] @ w + b).  Thread-per-row mat-vec.
// ---------------------------------------------------------------------------
__global__ void __launch_bounds__(256)
dense3_sigmoid(const float* __restrict__ X, const float* __restrict__ w,
               const float* __restrict__ b, float* __restrict__ out, int M) {
    const int r = blockIdx.x * blockDim.x + threadIdx.x;
    if (r >= M) return;
    const float4* xr = (const float4*)(X + (size_t)r * 128);
    const float4* wv = (const float4*)w;
    float s = 0.f;
    #pragma unroll
    for (int i = 0; i < 32; ++i) {
        const float4 a = xr[i], c = wv[i];
        s += a.x * c.x + a.y * c.y + a.z * c.z + a.w * c.w;
    }
    s += b[0];
    out[r] = 1.f / (1.f + __expf(-s));
}

// ---------------------------------------------------------------------------
extern "C" void kernel_launch(void* const* d_in, const int* in_sizes, int n_in,
                              void* d_out, int out_size, void* d_ws, size_t ws_size,
                              hipStream_t stream) {
    const float* x       = (const float*)d_in[0];
    const int*   snd     = (const int*)  d_in[1];
    const int*   rcv     = (const int*)  d_in[2];
    const float* W1      = (const float*)d_in[3];
    const float* b1      = (const float*)d_in[4];
    const float* gamma   = (const float*)d_in[5];
    const float* beta    = (const float*)d_in[6];
    const float* bn_mean = (const float*)d_in[7];
    const float* bn_var  = (const float*)d_in[8];
    const float* W_d1    = (const float*)d_in[9];
    const float* b_d1    = (const float*)d_in[10];
    const float* W_d2    = (const float*)d_in[11];
    const float* b_d2    = (const float*)d_in[12];
    const float* W_d3    = (const float*)d_in[13];
    const float* b_d3    = (const float*)d_in[14];

    const int nNodes = in_sizes[0] / CH;   // 409600
    const int nE     = in_sizes[1];        // 6,553,600
    const int Bg     = nNodes / GS;        // 4096

    char* ws = (char*)d_ws;
    float* hbuf = (float*)ws;                                    // [nNodes,64] agg->h in place
    const size_t hBytes = (size_t)nNodes * CH * sizeof(float);   // 104.9 MB
    float* y1   = (float*)(ws + hBytes);                         // [Bg,256]
    float* y2   = y1 + (size_t)Bg * 256;                         // [Bg,128]
    float* Wd1p = y2 + (size_t)Bg * 128;                         // repacked W_d1, 6.55 MB
    float* Wd2p = Wd1p + (size_t)K1 * 256;                       // repacked W_d2, 128 KB

    (void)hipMemsetAsync(hbuf, 0, hBytes, stream);
    repack_pairs  <<<(K1 / 2) * 256 / 256, 256, 0, stream>>>(W_d1, Wd1p, 256);
    repack_pairs  <<<128 * 128 / 256,      256, 0, stream>>>(W_d2, Wd2p, 128);
    edge_scatter  <<<8192,                 256, 0, stream>>>(x, snd, rcv, hbuf, nE);
    gemm1_bn_elu  <<<nNodes / (16 * 8),    256, 0, stream>>>(hbuf, W1, b1, gamma, beta, bn_mean, bn_var);
    dense1_relu   <<<dim3(Bg / 64, 4),     128, 0, stream>>>(hbuf, Wd1p, b_d1, y1);
    dense2_relu   <<<Bg / 16,              256, 0, stream>>>(y1, Wd2p, b_d2, y2);
    dense3_sigmoid<<<(Bg + 255) / 256,     256, 0, stream>>>(y2, W_d3, b_d3, (float*)d_out, Bg);
}